// MyGAT_26293789786473
// MI455X (gfx1250) — compile-verified
//
#include <hip/hip_runtime.h>
#include <hip/hip_bf16.h>
#include <math.h>

// ---------------------------------------------------------------------------
// GATv2 forward on MI455X (gfx1250).
//  Phase 1: x_src = x @ W_src, x_dst = x @ W_dst  via V_WMMA_F32_16X16X4_F32
//           (32 rows/block: each B fragment feeds 2 WMMAs; A staged in LDS)
//  Phase 2: per-edge logits (leaky_relu + att dot), segment max (CAS float max)
//  Phase 3: ex = exp(logit - seg_max[dst]); denom scatter-add
//  Phase 4: out[dst] += (ex/denom) * x_src[src]  via global_atomic_add_f32
// ---------------------------------------------------------------------------

typedef float v2f __attribute__((ext_vector_type(2)));
typedef float v8f __attribute__((ext_vector_type(8)));

#define F_IN 256
#define HC   128     // HEADS * C_OUT
#define C_OUT 64
#define NEG_SLOPE 0.2f
#define LDS_STRIDE 260   // 256 + 4 pad: lane l hits bank 4*l (conflict-free)

// ---------------- float atomic max via CAS (safe on all toolchains) --------
__device__ inline void atomicMaxF(float* addr, float val) {
    unsigned int* ua = reinterpret_cast<unsigned int*>(addr);
    unsigned int cur = *ua;
    while (__uint_as_float(cur) < val) {
        unsigned int prev = atomicCAS(ua, cur, __float_as_uint(val));
        if (prev == cur) break;
        cur = prev;
    }
}

// ---------------- init: out = bias, seg_max = -inf, denom = 0 --------------
__global__ __launch_bounds__(256)
void gat_init(float* __restrict__ out, const float* __restrict__ bias,
              float* __restrict__ seg_max, float* __restrict__ denom,
              long n_out, long n_seg) {
    long idx = (long)blockIdx.x * 256 + threadIdx.x;
    if (idx < n_out) out[idx] = bias[idx & (HC - 1)];
    if (idx < n_seg) { seg_max[idx] = -INFINITY; denom[idx] = 0.0f; }
}

// ---------------- WMMA fp32 GEMM: Y[N,128] = X[N,256] @ W[256,128] ---------
// Block = 256 threads = 8 waves; block owns 32 rows (two 16-row tiles).
// Wave w owns 16-col tile; each B fragment is reused by both row tiles.
__global__ __launch_bounds__(256)
void gat_gemm_wmma(const float* __restrict__ X, const float* __restrict__ W,
                   float* __restrict__ Y, int N) {
    __shared__ float As[32 * LDS_STRIDE];

    const int lane = threadIdx.x & 31;
    const int wave = threadIdx.x >> 5;
    const int row0 = blockIdx.x * 32;
    const int half = lane >> 4;        // 0: lanes 0-15 (K=ka,ka+1); 1: lanes 16-31 (K=ka+2,ka+3)
    const int l15  = lane & 15;
    const int bcol = wave * 16 + l15;  // 8 waves cover 128 output columns

    // cooperative load of 32x256 A tile into LDS (each x row read from HBM once)
    for (int i = threadIdx.x; i < 32 * F_IN; i += 256) {
        int r = i >> 8;                // row within tile
        int c = i & 255;
        int gr = row0 + r;
        if (gr >= N) gr = N - 1;       // clamp (N=100000 -> 3125 full blocks anyway)
        As[r * LDS_STRIDE + c] = X[(long)gr * F_IN + c];
    }
    __syncthreads();

    v8f acc0 = {}, acc1 = {};
    const float* a0p = &As[l15 * LDS_STRIDE];
    const float* a1p = &As[(16 + l15) * LDS_STRIDE];

    #pragma unroll 4
    for (int k = 0; k < F_IN; k += 4) {
        const int ka = k + half * 2;
        v2f a0, a1, b;
        // B 4x16 fragment: lane l15 = col, VGPR0/1 = K=ka,ka+1 (rows striped across lanes)
        b.x = W[(long)ka * HC + bcol];
        b.y = W[(long)(ka + 1) * HC + bcol];
        // A 16x4 fragments for the two row tiles
        a0.x = a0p[ka];  a0.y = a0p[ka + 1];
        a1.x = a1p[ka];  a1.y = a1p[ka + 1];
        acc0 = __builtin_amdgcn_wmma_f32_16x16x4_f32(false, a0, false, b,
                                                     (short)0, acc0, false, false);
        acc1 = __builtin_amdgcn_wmma_f32_16x16x4_f32(false, a1, false, b,
                                                     (short)0, acc1, false, false);
    }

    // C/D layout: VGPR v -> M = v + 8*half, N = l15
    const int rbase = row0 + half * 8;
    if (row0 + 32 <= N) {              // fast path: whole tile in bounds (no exec juggling)
        #pragma unroll
        for (int v = 0; v < 8; ++v) {
            Y[(long)(rbase + v) * HC + bcol]      = acc0[v];
            Y[(long)(rbase + 16 + v) * HC + bcol] = acc1[v];
        }
    } else {
        #pragma unroll
        for (int v = 0; v < 8; ++v) {
            int r0 = rbase + v, r1 = rbase + 16 + v;
            if (r0 < N) Y[(long)r0 * HC + bcol] = acc0[v];
            if (r1 < N) Y[(long)r1 * HC + bcol] = acc1[v];
        }
    }
}

// ---------------- per-edge logits + segment max (one wave32 per edge) ------
__global__ __launch_bounds__(256)
void gat_edge_logits(const float* __restrict__ xs, const float* __restrict__ xd,
                     const int* __restrict__ src, const int* __restrict__ dst,
                     const float* __restrict__ att,
                     float* __restrict__ logits, float* __restrict__ seg_max,
                     int E) {
    const int e = blockIdx.x * 8 + (threadIdx.x >> 5);
    if (e >= E) return;
    const int lane = threadIdx.x & 31;
    const int j = src[e];
    const int i = dst[e];
    const float* pj = xs + (long)j * HC;
    const float* pi = xd + (long)i * HC;

    #pragma unroll
    for (int h = 0; h < 2; ++h) {
        float s = 0.0f;
        #pragma unroll
        for (int t = 0; t < 2; ++t) {
            const int c = h * C_OUT + t * 32 + lane;
            float v = pj[c] + pi[c];
            v = (v > 0.0f) ? v : NEG_SLOPE * v;
            s += v * att[c];
        }
        // wave32 butterfly reduction
        #pragma unroll
        for (int m = 16; m > 0; m >>= 1) s += __shfl_xor(s, m, 32);
        if (lane == 0) {
            logits[(long)e * 2 + h] = s;
            atomicMaxF(&seg_max[(long)i * 2 + h], s);
        }
    }
}

// ---------------- ex = exp(logit - seg_max[dst]); denom += ex --------------
// One thread per (edge, head); ex overwrites logits in place.
__global__ __launch_bounds__(256)
void gat_edge_exp(float* __restrict__ logits_ex,
                  const int* __restrict__ dst,
                  const float* __restrict__ seg_max,
                  float* __restrict__ denom, int E) {
    const long idx = (long)blockIdx.x * 256 + threadIdx.x;
    if (idx >= (long)E * 2) return;
    const int e = (int)(idx >> 1);
    const int h = (int)(idx & 1);
    const int i = dst[e];
    const float ex = __expf(logits_ex[idx] - seg_max[(long)i * 2 + h]);
    logits_ex[idx] = ex;
    atomicAdd(&denom[(long)i * 2 + h], ex);
}

// ---------------- out[dst] += alpha * x_src[src] (one wave32 per edge) -----
__global__ __launch_bounds__(256)
void gat_scatter(const float* __restrict__ xs,
                 const int* __restrict__ src, const int* __restrict__ dst,
                 const float* __restrict__ ex, const float* __restrict__ denom,
                 float* __restrict__ out, int E) {
    const int e = blockIdx.x * 8 + (threadIdx.x >> 5);
    if (e >= E) return;
    const int lane = threadIdx.x & 31;
    const int j = src[e];
    const int i = dst[e];
    const float a0 = ex[(long)e * 2 + 0] / (denom[(long)i * 2 + 0] + 1e-16f);
    const float a1 = ex[(long)e * 2 + 1] / (denom[(long)i * 2 + 1] + 1e-16f);
    const float* pj = xs + (long)j * HC;
    float* po = out + (long)i * HC;
    atomicAdd(&po[lane],       a0 * pj[lane]);
    atomicAdd(&po[32 + lane],  a0 * pj[32 + lane]);
    atomicAdd(&po[64 + lane],  a1 * pj[64 + lane]);
    atomicAdd(&po[96 + lane],  a1 * pj[96 + lane]);
}

// ---------------------------------------------------------------------------
extern "C" void kernel_launch(void* const* d_in, const int* in_sizes, int n_in,
                              void* d_out, int out_size, void* d_ws, size_t ws_size,
                              hipStream_t stream) {
    const float* x     = (const float*)d_in[0];              // [N, 256]
    const int*   eidx  = (const int*)d_in[1];                // [2, E] (harness: integer -> int32)
    const float* W_src = (const float*)d_in[2];              // [256, 128]
    const float* W_dst = (const float*)d_in[3];              // [256, 128]
    const float* att   = (const float*)d_in[4];              // [2, 64]
    const float* bias  = (const float*)d_in[5];              // [128]

    const int N = in_sizes[0] / F_IN;
    const int E = in_sizes[1] / 2;
    const int* src = eidx;
    const int* dst = eidx + E;

    // workspace layout (floats)
    float* ws      = (float*)d_ws;
    float* xs      = ws;                          // [N,128]
    float* xd      = xs + (long)N * HC;           // [N,128]
    float* logits  = xd + (long)N * HC;           // [E,2]  (becomes ex in place)
    float* seg_max = logits + (long)E * 2;        // [N,2]
    float* denom   = seg_max + (long)N * 2;       // [N,2]

    const long n_out = (long)N * HC;
    const long n_seg = (long)N * 2;

    // 1) init
    {
        dim3 g((unsigned)((n_out + 255) / 256)), b(256);   // n_out >= n_seg
        gat_init<<<g, b, 0, stream>>>((float*)d_out, bias, seg_max, denom, n_out, n_seg);
    }
    // 2) two WMMA GEMMs (32 rows per block)
    {
        dim3 g((unsigned)((N + 31) / 32)), b(256);
        gat_gemm_wmma<<<g, b, 0, stream>>>(x, W_src, xs, N);
        gat_gemm_wmma<<<g, b, 0, stream>>>(x, W_dst, xd, N);
    }
    // 3) edge logits + segment max
    {
        dim3 g((unsigned)((E + 7) / 8)), b(256);
        gat_edge_logits<<<g, b, 0, stream>>>(xs, xd, src, dst, att, logits, seg_max, E);
    }
    // 4) exp + denom
    {
        dim3 g((unsigned)(((long)E * 2 + 255) / 256)), b(256);
        gat_edge_exp<<<g, b, 0, stream>>>(logits, dst, seg_max, denom, E);
    }
    // 5) normalize + scatter-add
    {
        dim3 g((unsigned)((E + 7) / 8)), b(256);
        gat_scatter<<<g, b, 0, stream>>>(xs, src, dst, logits, denom, (float*)d_out, E);
    }
}